// SA_44367012168191
// MI455X (gfx1250) — compile-verified
//
#include <hip/hip_runtime.h>

typedef float v2f __attribute__((ext_vector_type(2)));
typedef float v8f __attribute__((ext_vector_type(8)));
typedef unsigned int u32x4 __attribute__((ext_vector_type(4)));
typedef int i32x4 __attribute__((ext_vector_type(4)));
typedef int i32x8 __attribute__((ext_vector_type(8)));

#define WMMA_F32(a, b, c) \
  __builtin_amdgcn_wmma_f32_16x16x4_f32(false, (a), false, (b), (short)0, (c), false, false)

static constexpr int kB = 8;
static constexpr int kN = 4096;
static constexpr int kD = 256;
static constexpr int kE = 768;  // 3*D

__device__ __forceinline__ float rowmax16(float x) {
  x = fmaxf(x, __shfl_xor(x, 1, 32));
  x = fmaxf(x, __shfl_xor(x, 2, 32));
  x = fmaxf(x, __shfl_xor(x, 4, 32));
  x = fmaxf(x, __shfl_xor(x, 8, 32));
  return x;
}
__device__ __forceinline__ float rowsum16(float x) {
  x += __shfl_xor(x, 1, 32);
  x += __shfl_xor(x, 2, 32);
  x += __shfl_xor(x, 4, 32);
  x += __shfl_xor(x, 8, 32);
  return x;
}

#if __has_builtin(__builtin_amdgcn_tensor_load_to_lds)
#define USE_TDM 1
// Issue one 2D TDM tile load: tile (tile1 rows x tile0 elems) of an f32
// tensor (dim1 rows x dim0 elems, row stride stride0), into LDS at lds_addr.
// g1w0 carries {data_size=4B, pad_*} control bits (group1 word 0).
// Descriptor packing per CDNA5 ISA ch.8 (D# groups; type=2 "image").
// Toolchain here is the 6-arg clang-23 variant:
//   (u32x4 g0, i32x8 g1, i32x4 g2, i32x4 g3, i32x8 extra, i32 cpol)
__device__ __forceinline__ void tdm_load_2d(unsigned lds_addr,
                                            const float* gptr, unsigned dim0,
                                            unsigned dim1, unsigned tile0,
                                            unsigned tile1, unsigned stride0,
                                            unsigned g1w0) {
  const unsigned long long ga = (unsigned long long)(uintptr_t)gptr;
  u32x4 g0;
  g0[0] = 1u;                                             // count=1, no gather
  g0[1] = lds_addr;                                       // lds_addr [63:32]
  g0[2] = (unsigned)(ga & 0xFFFFFFFFu);                   // global_addr lo
  g0[3] = (unsigned)((ga >> 32) & 0x01FFFFFFu) | (2u << 30);  // ga hi | type=2
  i32x8 g1;
  g1[0] = (int)g1w0;                             // wg_mask=0|data_size|pad ctl
  g1[1] = (int)((dim0 & 0xFFFFu) << 16);         // atomic_addr=0 | dim0 lo16
  g1[2] = (int)((dim0 >> 16) | ((dim1 & 0xFFFFu) << 16));  // dim0 hi | dim1 lo
  g1[3] = (int)((dim1 >> 16) | (tile0 << 16));   // dim1 hi | tile_dim0
  g1[4] = (int)(tile1 & 0xFFFFu);                // tile_dim1 | tile_dim2=0
  g1[5] = (int)stride0;                          // tensor_dim0_stride lo32
  g1[6] = 0;                                     // stride0 hi16 | stride1 lo16
  g1[7] = 0;                                     // stride1 hi32
  const i32x4 z4 = {0, 0, 0, 0};                 // groups 2/3: 2D tensor
  const i32x8 z8 = {0, 0, 0, 0, 0, 0, 0, 0};
  __builtin_amdgcn_tensor_load_to_lds(g0, g1, z4, z4, z8, 0);
}
#else
#define USE_TDM 0
#endif

// ---------------------------------------------------------------------------
// Kernel 1: qkv = x @ W^T + b  (fp32 WMMA 16x16x4).
// Q is written pre-scaled by 1/sqrt(D); K is written transposed [B][D][N].
// grid = (B*N/16, 3D/64), block = 128 (4 waves, one 16x16 col tile each).
// ---------------------------------------------------------------------------
__global__ __launch_bounds__(128) void qkv_proj_kernel(
    const float* __restrict__ x, const float* __restrict__ W,
    const float* __restrict__ bias, float* __restrict__ q,
    float* __restrict__ kt, float* __restrict__ v) {
  const int lane = threadIdx.x & 31;
  const int wave = threadIdx.x >> 5;
  const int hi = lane >> 4;  // lane half: K-pair select per ISA 16x4 layout
  const int l15 = lane & 15;
  const long r0 = (long)blockIdx.x * 16;       // row tile in flattened B*N
  const int e0 = blockIdx.y * 64 + wave * 16;  // output-column tile
  const int ecol = e0 + l15;

  const float* xrow = x + (r0 + l15) * (long)kD;  // A row for this lane
  const float* wrow = W + (long)ecol * kD;        // B col = W row (x @ W^T)

  v8f acc = {};
#pragma unroll 8
  for (int k0 = 0; k0 < kD; k0 += 4) {
    const int ka = k0 + 2 * hi;
    v2f a, bm;
    a.x = xrow[ka];
    a.y = xrow[ka + 1];
    bm.x = wrow[ka];
    bm.y = wrow[ka + 1];
    acc = WMMA_F32(a, bm, acc);
  }

  const float bv = bias[ecol];
  const int seg = e0 >> 8;  // 0=Q, 1=K, 2=V (wave tile never straddles)
#pragma unroll
  for (int g = 0; g < 8; ++g) {
    const long rr = r0 + g + 8 * hi;  // C layout: VGPR g -> row g (+8 hi half)
    const float val = acc[g] + bv;
    if (seg == 0) {
      q[rr * kD + ecol] = val * 0.0625f;  // fold 1/sqrt(256) into Q
    } else if (seg == 1) {
      const long bb = rr >> 12;
      const long n = rr & (kN - 1);
      kt[(bb * kD + (ecol - 256)) * (long)kN + n] = val;  // transposed K
    } else {
      v[rr * kD + (ecol - 512)] = val;
    }
  }
}

// ---------------------------------------------------------------------------
// Kernel 2: flash attention. block = 256 (8 waves); each wave owns 16 query
// rows. The 16-key K^T-tile and V-tile are staged into LDS by the Tensor
// Data Mover (wave 0 issues 2 TDM loads, waits TENSORcnt, barrier publishes).
// The V tile uses the TDM pad feature (4 DWORDs every 256) -> stride-260
// rows, keeping the P@V B-fragment reads bank-conflict free.
// Online softmax with per-row running max/sum; P goes through per-wave LDS
// scratch to transpose C-layout -> A-fragment layout for the P@V WMMAs.
// grid = (N/128, B).
// ---------------------------------------------------------------------------
__global__ __launch_bounds__(256) void attn_kernel(
    const float* __restrict__ q, const float* __restrict__ kt,
    const float* __restrict__ v, float* __restrict__ out) {
  __shared__ float sK[kD * 16];     // [d][key], stride 16 (conflict-free)
  __shared__ float sV[16 * 260];    // [key][d], padded stride 260
  __shared__ float sP[8][16 * 17];  // per-wave P scratch, padded stride 17

  const int tid = threadIdx.x;
  const int lane = tid & 31;
  const int wave = tid >> 5;
  const int hi = lane >> 4;
  const int l15 = lane & 15;
  const int b = blockIdx.y;
  const int q0 = blockIdx.x * 128 + wave * 16;

  const float* qrow = q + ((long)b * kN + q0 + l15) * kD;  // A row per lane
  const float* ktb = kt + (long)b * kD * kN;
  const float* vb = v + (long)b * kN * kD;

  v8f o[16] = {};  // 16 d-tiles x v8f : full 16x256 output accumulator
  float mrow[8], lrow[8];
#pragma unroll
  for (int g = 0; g < 8; ++g) {
    mrow[g] = -3.0e38f;
    lrow[g] = 0.0f;
  }

  for (int key0 = 0; key0 < kN; key0 += 16) {
    __syncthreads();  // previous tile fully consumed
#if USE_TDM
    if (wave == 0) {
      // K^T tile: tensor [256][4096], tile 256 rows x 16 keys -> sK [d][key]
      tdm_load_2d((unsigned)(uintptr_t)sK, ktb + key0, kN, kD, 16, kD, kN,
                  /*data_size=4B*/ 0x20000u);
      // V tile: tensor [4096][256], tile 16 rows x 256 d -> sV, padded:
      // pad_enable | pad_interval=7 (256 dw) | pad_amount=3 (4 dw)
      tdm_load_2d((unsigned)(uintptr_t)sV, vb + (long)key0 * kD, kD, kN, kD,
                  16, kD, 0x20000u | (1u << 20) | (7u << 22) | (3u << 25));
      __builtin_amdgcn_s_wait_tensorcnt((short)0);
    }
#else
#pragma unroll
    for (int i = 0; i < 16; ++i) {
      const int li = i * 256 + tid;
      sK[li] = ktb[(long)(li >> 4) * kN + key0 + (li & 15)];
    }
#pragma unroll
    for (int i = 0; i < 16; ++i) {
      sV[i * 260 + tid] = vb[(long)(key0 + i) * kD + tid];
    }
#endif
    __syncthreads();  // tile visible to all 8 waves

    // S = Qs @ K^T  (scale already folded into Q)
    v8f s = {};
#pragma unroll 8
    for (int d0 = 0; d0 < kD; d0 += 4) {
      const int ka = d0 + 2 * hi;
      v2f a, bm;
      a.x = qrow[ka];
      a.y = qrow[ka + 1];
      bm.x = sK[ka * 16 + l15];
      bm.y = sK[(ka + 1) * 16 + l15];
      s = WMMA_F32(a, bm, s);
    }

    // Online softmax: each lane holds 8 rows' worth (one col each)
    float* pw = &sP[wave][0];
    v8f av;
#pragma unroll
    for (int g = 0; g < 8; ++g) {
      const float mx = rowmax16(s[g]);
      const float mn = fmaxf(mrow[g], mx);
      const float alpha = __expf(mrow[g] - mn);
      const float p = __expf(s[g] - mn);
      const float rs = rowsum16(p);
      lrow[g] = lrow[g] * alpha + rs;
      mrow[g] = mn;
      av[g] = alpha;
      pw[(g + 8 * hi) * 17 + l15] = p;  // C-layout -> row-major P in LDS
    }
#pragma unroll
    for (int t = 0; t < 16; ++t) o[t] *= av;

    // O += P @ V  (A-frags of P read back from per-wave LDS; same-wave DS
    // ordering guarantees the transpose round-trip)
#pragma unroll
    for (int kk0 = 0; kk0 < 16; kk0 += 4) {
      const int kb2 = kk0 + 2 * hi;
      v2f a;
      a.x = pw[l15 * 17 + kb2];
      a.y = pw[l15 * 17 + kb2 + 1];
#pragma unroll
      for (int t = 0; t < 16; ++t) {
        v2f bm;
        bm.x = sV[kb2 * 260 + t * 16 + l15];
        bm.y = sV[(kb2 + 1) * 260 + t * 16 + l15];
        o[t] = WMMA_F32(a, bm, o[t]);
      }
    }
  }

  // Epilogue: normalize by running sum, store C-layout tiles
  float* ob = out + ((long)b * kN + q0) * kD;
#pragma unroll
  for (int t = 0; t < 16; ++t) {
#pragma unroll
    for (int g = 0; g < 8; ++g) {
      ob[(long)(g + 8 * hi) * kD + t * 16 + l15] = o[t][g] / lrow[g];
    }
  }
}

// ---------------------------------------------------------------------------
extern "C" void kernel_launch(void* const* d_in, const int* in_sizes, int n_in,
                              void* d_out, int out_size, void* d_ws,
                              size_t ws_size, hipStream_t stream) {
  const float* x = (const float*)d_in[0];     // [B,N,D]
  const float* W = (const float*)d_in[1];     // [3D,D]
  const float* bias = (const float*)d_in[2];  // [3D]
  float* out = (float*)d_out;                 // [B,N,D]

  const size_t per = (size_t)kB * kN * kD;
  float* qws = (float*)d_ws;  // Q pre-scaled   [B,N,D]
  float* ktws = qws + per;    // K transposed   [B,D,N]
  float* vws = ktws + per;    // V              [B,N,D]

  dim3 g1((kB * kN) / 16, kE / 64);
  qkv_proj_kernel<<<g1, dim3(128), 0, stream>>>(x, W, bias, qws, ktws, vws);

  dim3 g2(kN / 128, kB);
  attn_kernel<<<g2, dim3(256), 0, stream>>>(qws, ktws, vws, out);
}